// GraphSAGE_79671643341335
// MI455X (gfx1250) — compile-verified
//
#include <hip/hip_runtime.h>

typedef __attribute__((ext_vector_type(2))) float v2f;
typedef __attribute__((ext_vector_type(8))) float v8f;

#define N_NODES  100000
#define N_EDGES  3200000
#define N_GRAPHS 256
#define IN_CH    11
#define EDGE_CH  22
#define HID      64
#define N_CLS    2

// ---------------------------------------------------------------- zero fill
__global__ void k_zero(float* __restrict__ p, long long n) {
  long long i = (long long)blockIdx.x * blockDim.x + threadIdx.x;
  if (i < n) p[i] = 0.0f;
}

// ------------------------------------------------- copy x into padded [N,16]
__global__ void k_init_xin(const float* __restrict__ x, float* __restrict__ xin) {
  int i = blockIdx.x * blockDim.x + threadIdx.x;   // over N*16
  if (i >= N_NODES * 16) return;
  int n = i >> 4, c = i & 15;
  xin[i] = (c < IN_CH) ? x[n * IN_CH + c] : 0.0f;
}

// --------------- per-edge correction ec = ea @ we.T + be, scatter to src;
// --------------- also builds the degree vector (deg[dst] += 1) once.
__global__ void k_edge_scatter(const float* __restrict__ ea,
                               const int*   __restrict__ src,
                               const int*   __restrict__ dst,
                               const float* __restrict__ we,
                               const float* __restrict__ be,
                               float* __restrict__ xin,
                               float* __restrict__ deg) {
  __shared__ float swe[IN_CH * EDGE_CH];
  __shared__ float sbe[IN_CH];
  for (int i = threadIdx.x; i < IN_CH * EDGE_CH; i += blockDim.x) swe[i] = we[i];
  for (int i = threadIdx.x; i < IN_CH;           i += blockDim.x) sbe[i] = be[i];
  __syncthreads();
  int e = blockIdx.x * blockDim.x + threadIdx.x;
  if (e >= N_EDGES) return;
  float a[EDGE_CH];
#pragma unroll
  for (int j = 0; j < EDGE_CH; ++j) a[j] = ea[(long long)e * EDGE_CH + j];
  int s = src[e];
#pragma unroll
  for (int c = 0; c < IN_CH; ++c) {
    float acc = sbe[c];
#pragma unroll
    for (int j = 0; j < EDGE_CH; ++j) acc = fmaf(swe[c * EDGE_CH + j], a[j], acc);
    atomicAdd(&xin[(long long)s * 16 + c], acc);
  }
  atomicAdd(&deg[dst[e]], 1.0f);
}

// ----------------- segment-sum: agg[dst] += feat[src], one thread per
// ----------------- (edge, channel) -> coalesced loads, L2-resident atomics.
__global__ void k_aggregate(const float* __restrict__ feat, int stride, int cin,
                            const int* __restrict__ src, const int* __restrict__ dst,
                            float* __restrict__ agg, long long total) {
  long long i = (long long)blockIdx.x * blockDim.x + threadIdx.x;
  if (i >= total) return;
  int       c = (int)(i % stride);
  long long e = i / stride;
  if (c >= cin) return;
  int s = src[e], d = dst[e];
  atomicAdd(&agg[(long long)d * stride + c], feat[(long long)s * stride + c]);
}

// --------- node update: out = relu( (agg/deg) @ lw.T + lb + xin @ rw.T )
// --------- 256 thr = 8 waves; each wave owns one 16x16 f32 WMMA tile.
// --------- Block tile: 32 nodes x 64 out-channels. Safe for out==xin
// --------- (inputs staged to LDS before any store).
__global__ void __launch_bounds__(256)
k_node_update(const float* __restrict__ xin, int xstride,
              const float* __restrict__ agg, int astride,
              const float* __restrict__ deg,
              const float* __restrict__ lw,   // [HID, cin]
              const float* __restrict__ rw,   // [HID, cin]
              const float* __restrict__ lb,   // [HID]
              int cin, int kpad,              // kpad: cin padded to mult of 4
              float* __restrict__ out) {      // [N, 64]
  __shared__ float s_mean[32][65];
  __shared__ float s_x[32][65];
  __shared__ float s_lwT[64][65];   // [k][n] = lw[n][k]
  __shared__ float s_rwT[64][65];

  const int tid   = threadIdx.x;
  const int node0 = blockIdx.x * 32;

  // stage transposed weights (zero-pad k >= cin)
  for (int i = tid; i < kpad * HID; i += 256) {
    int k = i / HID, n = i % HID;
    float wl = (k < cin) ? lw[n * cin + k] : 0.0f;
    float wr = (k < cin) ? rw[n * cin + k] : 0.0f;
    s_lwT[k][n] = wl;
    s_rwT[k][n] = wr;
  }
  // stage node features: mean = agg/max(deg,1), and root x
  for (int i = tid; i < 32 * kpad; i += 256) {
    int m = i & 31, k = i >> 5;
    int node = node0 + m;
    float dv = fmaxf(deg[node], 1.0f);
    s_mean[m][k] = (k < cin) ? agg[(long long)node * astride + k] / dv : 0.0f;
    s_x[m][k]    = (k < cin) ? xin[(long long)node * xstride + k]      : 0.0f;
  }
  __syncthreads();

  const int lane  = tid & 31;          // wave32
  const int w     = tid >> 5;          // wave id 0..7
  const int mtile = (w >> 2) * 16;     // 0 / 16
  const int ntile = (w & 3)  * 16;     // 0..48
  const int half  = lane >> 4;         // K-half select for A/B fragments
  const int mloc  = mtile + (lane & 15);
  const int nloc  = ntile + (lane & 15);

  // C/D init = bias broadcast over M
  v8f acc;
  {
    float b = lb[nloc];
#pragma unroll
    for (int r = 0; r < 8; ++r) acc[r] = b;
  }

  // dual WMMA accumulation chains: mean@lw.T and x@rw.T
  for (int k = 0; k < kpad; k += 4) {
    int ka = k + half * 2;
    v2f a, b;
    a.x = s_mean[mloc][ka];  a.y = s_mean[mloc][ka + 1];
    b.x = s_lwT[ka][nloc];   b.y = s_lwT[ka + 1][nloc];
    acc = __builtin_amdgcn_wmma_f32_16x16x4_f32(false, a, false, b,
                                                (short)0, acc, false, false);
    a.x = s_x[mloc][ka];     a.y = s_x[mloc][ka + 1];
    b.x = s_rwT[ka][nloc];   b.y = s_rwT[ka + 1][nloc];
    acc = __builtin_amdgcn_wmma_f32_16x16x4_f32(false, a, false, b,
                                                (short)0, acc, false, false);
  }

  // ReLU + store; C/D layout: VGPR r -> M=r (lanes 0-15) / M=r+8 (lanes 16-31)
#pragma unroll
  for (int r = 0; r < 8; ++r) {
    int m = mtile + r + half * 8;
    out[(long long)(node0 + m) * 64 + nloc] = fmaxf(acc[r], 0.0f);
  }
}

// ------------------------------------------------- global mean pool (sums)
__global__ void k_pool(const float* __restrict__ h, const int* __restrict__ batch,
                       float* __restrict__ psum, float* __restrict__ cnt) {
  long long i = (long long)blockIdx.x * blockDim.x + threadIdx.x;
  if (i >= (long long)N_NODES * 64) return;
  int c = (int)(i & 63);
  int n = (int)(i >> 6);
  int b = batch[n];
  atomicAdd(&psum[b * 64 + c], h[i]);
  if (c == 0) atomicAdd(&cnt[b], 1.0f);
}

// ------------------------------------------------- classifier [256,64]x[64,2]
__global__ void k_final(const float* __restrict__ psum, const float* __restrict__ cnt,
                        const float* __restrict__ wf, const float* __restrict__ bf,
                        float* __restrict__ out) {
  int i = blockIdx.x * blockDim.x + threadIdx.x;
  if (i >= N_GRAPHS * N_CLS) return;
  int g = i / N_CLS, k = i % N_CLS;
  float inv = 1.0f / fmaxf(cnt[g], 1.0f);
  float acc = bf[k];
#pragma unroll
  for (int c = 0; c < HID; ++c)
    acc = fmaf(psum[g * 64 + c] * inv, wf[k * HID + c], acc);
  out[i] = acc;
}

extern "C" void kernel_launch(void* const* d_in, const int* in_sizes, int n_in,
                              void* d_out, int out_size, void* d_ws, size_t ws_size,
                              hipStream_t stream) {
  (void)in_sizes; (void)n_in; (void)out_size; (void)ws_size;
  const float* x     = (const float*)d_in[0];
  const float* ea    = (const float*)d_in[1];
  const int*   ei    = (const int*)d_in[2];
  const int*   batch = (const int*)d_in[3];
  const float* we    = (const float*)d_in[4];
  const float* be    = (const float*)d_in[5];
  const float* l1w   = (const float*)d_in[6];
  const float* l1b   = (const float*)d_in[7];
  const float* r1w   = (const float*)d_in[8];
  const float* l2w   = (const float*)d_in[9];
  const float* l2b   = (const float*)d_in[10];
  const float* r2w   = (const float*)d_in[11];
  const float* l3w   = (const float*)d_in[12];
  const float* l3b   = (const float*)d_in[13];
  const float* r3w   = (const float*)d_in[14];
  const float* wf    = (const float*)d_in[15];
  const float* bf    = (const float*)d_in[16];
  const int* src = ei;
  const int* dst = ei + N_EDGES;

  // workspace carve-out (~58 MB)
  char* ws = (char*)d_ws;
  size_t off = 0;
  auto carve = [&](size_t bytes) {
    void* p = ws + off;
    off = (off + bytes + 255) & ~(size_t)255;
    return p;
  };
  float* xin  = (float*)carve(sizeof(float) * (size_t)N_NODES * 16);
  float* deg  = (float*)carve(sizeof(float) * (size_t)N_NODES);
  float* agg  = (float*)carve(sizeof(float) * (size_t)N_NODES * 64);
  float* h    = (float*)carve(sizeof(float) * (size_t)N_NODES * 64);
  float* psum = (float*)carve(sizeof(float) * (size_t)N_GRAPHS * 64);
  float* cnt  = (float*)carve(sizeof(float) * (size_t)N_GRAPHS);

  const int T = 256;
  auto blocks = [](long long n, int t) { return (unsigned)((n + t - 1) / t); };

  // edge correction + degree
  k_zero<<<blocks(N_NODES, T), T, 0, stream>>>(deg, N_NODES);
  k_init_xin<<<blocks((long long)N_NODES * 16, T), T, 0, stream>>>(x, xin);
  k_edge_scatter<<<blocks(N_EDGES, T), T, 0, stream>>>(ea, src, dst, we, be, xin, deg);

  // ---- layer 1 (cin=11, kpad=12, feature stride 16) -> h
  k_zero<<<blocks((long long)N_NODES * 16, T), T, 0, stream>>>(agg, (long long)N_NODES * 16);
  k_aggregate<<<blocks((long long)N_EDGES * 16, T), T, 0, stream>>>(
      xin, 16, IN_CH, src, dst, agg, (long long)N_EDGES * 16);
  k_node_update<<<N_NODES / 32, 256, 0, stream>>>(
      xin, 16, agg, 16, deg, l1w, r1w, l1b, IN_CH, 12, h);

  // ---- layer 2 (cin=64) in place on h
  k_zero<<<blocks((long long)N_NODES * 64, T), T, 0, stream>>>(agg, (long long)N_NODES * 64);
  k_aggregate<<<blocks((long long)N_EDGES * 64, T), T, 0, stream>>>(
      h, 64, HID, src, dst, agg, (long long)N_EDGES * 64);
  k_node_update<<<N_NODES / 32, 256, 0, stream>>>(
      h, 64, agg, 64, deg, l2w, r2w, l2b, HID, 64, h);

  // ---- layer 3 (cin=64) in place on h
  k_zero<<<blocks((long long)N_NODES * 64, T), T, 0, stream>>>(agg, (long long)N_NODES * 64);
  k_aggregate<<<blocks((long long)N_EDGES * 64, T), T, 0, stream>>>(
      h, 64, HID, src, dst, agg, (long long)N_EDGES * 64);
  k_node_update<<<N_NODES / 32, 256, 0, stream>>>(
      h, 64, agg, 64, deg, l3w, r3w, l3b, HID, 64, h);

  // ---- global mean pool + classifier
  k_zero<<<blocks(N_GRAPHS * 64, T), T, 0, stream>>>(psum, N_GRAPHS * 64);
  k_zero<<<blocks(N_GRAPHS, T), T, 0, stream>>>(cnt, N_GRAPHS);
  k_pool<<<blocks((long long)N_NODES * 64, T), T, 0, stream>>>(h, batch, psum, cnt);
  k_final<<<blocks(N_GRAPHS * N_CLS, T), T, 0, stream>>>(psum, cnt, wf, bf, (float*)d_out);
}